// ANRProtectedModule_81492709474815
// MI455X (gfx1250) — compile-verified
//
#include <hip/hip_runtime.h>
#include <math.h>

// ---------------------------------------------------------------------------
// Problem constants
// ---------------------------------------------------------------------------
#define B_   128
#define C_   3
#define H_   256
#define W_   256
#define HW_  (H_ * W_)          // 65536
#define NPLANE (B_ * C_)        // 384
#define NOUT 1000
#define KPAD 4                  // channel dim padded to 4 for WMMA K

// conv tiling
#define TS   64
#define HALO 3
#define LW   (TS + 2 * HALO)    // 70

// workspace layout (float offsets)
#define O_SUMS   0              // 384
#define O_CMIN   384            // 384
#define O_CMAX   768            // 384
#define O_BMIN   1152           // 128
#define O_BMAX   1280           // 128
#define O_MEANS  1408           // 128*4
#define O_HENT   1920           // 384
#define O_HS     2304           // 128
#define O_IV     2432           // 128
#define O_PART   2560           // 384*16 = 6144
#define O_NMEANS 8704           // 128*4
#define O_NLOG   9216           // 128*1000
#define O_CLS    137216         // 128 ints
#define O_NCLS   137344         // 128 ints
#define WS_FLOATS 137472

typedef __attribute__((ext_vector_type(2))) float v2f;
typedef __attribute__((ext_vector_type(8))) float v8f;

// ---------------------------------------------------------------------------
// K1: per-(b,c) plane sum / min / max   (grid = 384 planes, 256 thr)
// ---------------------------------------------------------------------------
__global__ void plane_stats_kernel(const float* __restrict__ x,
                                   float* __restrict__ sums,
                                   float* __restrict__ cmin,
                                   float* __restrict__ cmax) {
  const int plane = blockIdx.x;
  const int tid = threadIdx.x;
  const float4* p = (const float4*)(x + (size_t)plane * HW_);
  float s = 0.f, mn = INFINITY, mx = -INFINITY;
  for (int i = tid; i < HW_ / 4; i += 256) {
    float4 v = p[i];
    s += v.x + v.y + v.z + v.w;
    mn = fminf(mn, fminf(fminf(v.x, v.y), fminf(v.z, v.w)));
    mx = fmaxf(mx, fmaxf(fmaxf(v.x, v.y), fmaxf(v.z, v.w)));
  }
  __shared__ float ss[256], smn[256], smx[256];
  ss[tid] = s; smn[tid] = mn; smx[tid] = mx;
  __syncthreads();
  for (int st = 128; st > 0; st >>= 1) {
    if (tid < st) {
      ss[tid] += ss[tid + st];
      smn[tid] = fminf(smn[tid], smn[tid + st]);
      smx[tid] = fmaxf(smx[tid], smx[tid + st]);
    }
    __syncthreads();
  }
  if (tid == 0) { sums[plane] = ss[0]; cmin[plane] = smn[0]; cmax[plane] = smx[0]; }
}

// ---------------------------------------------------------------------------
// K2: per-image min/max + padded channel means   (128 threads)
// ---------------------------------------------------------------------------
__global__ void per_image_kernel(const float* __restrict__ sums,
                                 const float* __restrict__ cmin,
                                 const float* __restrict__ cmax,
                                 float* __restrict__ bmin,
                                 float* __restrict__ bmax,
                                 float* __restrict__ meansPad) {
  const int b = blockIdx.x * blockDim.x + threadIdx.x;
  if (b >= B_) return;
  bmin[b] = fminf(fminf(cmin[b * 3], cmin[b * 3 + 1]), cmin[b * 3 + 2]);
  bmax[b] = fmaxf(fmaxf(cmax[b * 3], cmax[b * 3 + 1]), cmax[b * 3 + 2]);
  const float inv = 1.0f / (float)HW_;
  meansPad[b * KPAD + 0] = sums[b * 3 + 0] * inv;
  meansPad[b * KPAD + 1] = sums[b * 3 + 1] * inv;
  meansPad[b * KPAD + 2] = sums[b * 3 + 2] * inv;
  meansPad[b * KPAD + 3] = 0.0f;
}

// ---------------------------------------------------------------------------
// K3/K9: linear head via V_WMMA_F32_16X16X4_F32.
//   D = A(16x4 means) x B(4x16 of W) + C(bias)
//   grid = (8 m-tiles, 63 n-tiles), block = 32 (one wave per 16x16 tile).
//   ISA layouts (7.12.2):
//     A 16x4 f32 : lane L -> M = L%16 ; VGPR{0,1} = K{k0,k0+1}, k0 = 2*(L/16)
//     B 4x16 f32 : lane L -> N = L%16 ; VGPR{0,1} = K{k0,k0+1}, k0 = 2*(L/16)
//     C/D 16x16  : lane L -> N = L%16 ; VGPR v -> M = v + 8*(L/16)
// ---------------------------------------------------------------------------
__global__ void wmma_head_kernel(const float* __restrict__ meansPad, // [128*4]
                                 const float* __restrict__ Wm,       // [3*1000]
                                 const float* __restrict__ bias,     // [1000]
                                 float* __restrict__ out) {          // [128*1000]
  const int lane = threadIdx.x;          // 0..31
  const int mt = blockIdx.x;             // 0..7
  const int nt = blockIdx.y;             // 0..62
  const int half = lane >> 4;            // 0/1
  const int l16 = lane & 15;
  const int k0 = half * 2;
  const int m = mt * 16 + l16;
  const int n = nt * 16 + l16;
  const bool nvalid = (n < NOUT);

  v2f a;
  a.x = meansPad[m * KPAD + k0];
  a.y = meansPad[m * KPAD + k0 + 1];

  v2f bm;
  bm.x = (nvalid && k0 < C_)     ? Wm[k0 * NOUT + n]       : 0.0f;
  bm.y = (nvalid && k0 + 1 < C_) ? Wm[(k0 + 1) * NOUT + n] : 0.0f;

  const float bn = nvalid ? bias[n] : 0.0f;
  v8f c;
#pragma unroll
  for (int v = 0; v < 8; ++v) c[v] = bn;  // C[m][n] = bias[n] for all m

  v8f d = __builtin_amdgcn_wmma_f32_16x16x4_f32(
      /*neg_a=*/false, a, /*neg_b=*/false, bm,
      /*c_mod=*/(short)0, c, /*reuse_a=*/false, /*reuse_b=*/false);

  if (nvalid) {
#pragma unroll
    for (int v = 0; v < 8; ++v) {
      const int mo = mt * 16 + v + half * 8;
      out[(size_t)mo * NOUT + n] = d[v];
    }
  }
}

// ---------------------------------------------------------------------------
// K4/K10: per-row argmax (first-max tie-break, matching jnp.argmax)
// ---------------------------------------------------------------------------
__global__ void argmax_kernel(const float* __restrict__ logits,
                              int* __restrict__ out, int N) {
  const int b = blockIdx.x;
  const int tid = threadIdx.x;
  const float* row = logits + (size_t)b * N;
  float best = -INFINITY;
  int bi = N;
  for (int n = tid; n < N; n += 256) {
    float v = row[n];
    if (v > best || (v == best && n < bi)) { best = v; bi = n; }
  }
  __shared__ float sv[256];
  __shared__ int si[256];
  sv[tid] = best; si[tid] = bi;
  __syncthreads();
  for (int st = 128; st > 0; st >>= 1) {
    if (tid < st) {
      if (sv[tid + st] > sv[tid] ||
          (sv[tid + st] == sv[tid] && si[tid + st] < si[tid])) {
        sv[tid] = sv[tid + st]; si[tid] = si[tid + st];
      }
    }
    __syncthreads();
  }
  if (tid == 0) out[b] = si[0];
}

// ---------------------------------------------------------------------------
// K5: per-(b,c) 256-bin histogram + Shannon entropy  (grid = 384, 256 thr)
// ---------------------------------------------------------------------------
__global__ void entropy_kernel(const float* __restrict__ x,
                               const float* __restrict__ bmin,
                               const float* __restrict__ bmax,
                               float* __restrict__ hent) {
  __shared__ unsigned hist[256];
  const int plane = blockIdx.x;
  const int b = plane / 3;
  const int tid = threadIdx.x;
  hist[tid] = 0u;
  __syncthreads();
  const float mn = bmin[b];
  const float scale = 255.0f / (bmax[b] - mn);
  const float4* p = (const float4*)(x + (size_t)plane * HW_);
  for (int i = tid; i < HW_ / 4; i += 256) {
    float4 v = p[i];
    float t0 = (v.x - mn) * scale, t1 = (v.y - mn) * scale;
    float t2 = (v.z - mn) * scale, t3 = (v.w - mn) * scale;
    int b0 = min(max((int)t0, 0), 255);
    int b1 = min(max((int)t1, 0), 255);
    int b2 = min(max((int)t2, 0), 255);
    int b3 = min(max((int)t3, 0), 255);
    atomicAdd(&hist[b0], 1u);
    atomicAdd(&hist[b1], 1u);
    atomicAdd(&hist[b2], 1u);
    atomicAdd(&hist[b3], 1u);
  }
  __syncthreads();
  const float pb = (float)hist[tid] * (1.0f / (float)HW_);
  float e = (pb > 0.0f) ? -pb * __log2f(pb) : 0.0f;
  __shared__ float red[256];
  red[tid] = e;
  __syncthreads();
  for (int st = 128; st > 0; st >>= 1) {
    if (tid < st) red[tid] += red[tid + st];
    __syncthreads();
  }
  if (tid == 0) hent[plane] = red[0];
}

// ---------------------------------------------------------------------------
// K6: per-image entropy mean -> interval  (128 threads)
// ---------------------------------------------------------------------------
__global__ void interval_kernel(const float* __restrict__ hent,
                                float* __restrict__ hs,
                                float* __restrict__ ivv) {
  const int b = blockIdx.x * blockDim.x + threadIdx.x;
  if (b >= B_) return;
  const float h = (hent[b * 3] + hent[b * 3 + 1] + hent[b * 3 + 2]) * (1.0f / 3.0f);
  hs[b] = h;
  ivv[b] = (h < 4.0f) ? 128.0f : ((h < 5.0f) ? 64.0f : 43.0f);
}

// ---------------------------------------------------------------------------
// K7: fused quantize -> 7x7 cross conv -> blend -> select -> partial mean.
//     Never materializes quantx/smoothed/processed in global memory.
//     grid = (4, 4, 384) tiles, block = 256. LDS: 70x70 quant tile + 64x64
//     original tile + 256 reduce = ~37 KB (of 320 KB/WGP).
// ---------------------------------------------------------------------------
__global__ void fused_process_kernel(const float* __restrict__ x,
                                     const float* __restrict__ bmin,
                                     const float* __restrict__ bmax,
                                     const float* __restrict__ ivv,
                                     float* __restrict__ partials) {
  __shared__ float q[LW * LW];       // quantized halo tile (zero-padded)
  __shared__ float og[TS * TS];      // original interior pixels
  __shared__ float red[256];

  const int plane = blockIdx.z;
  const int b = plane / 3;
  const int tx = blockIdx.x, ty = blockIdx.y;
  const int tid = threadIdx.x;
  const float* p = x + (size_t)plane * HW_;

  const float mn = bmin[b];
  const float range = bmax[b] - mn;
  const float scale = 255.0f / range;
  const float iv = ivv[b];
  const float inv_iv = 1.0f / iv;
  const float deq = iv * (1.0f / 255.0f) * range;
  const bool quant_only = (iv > 44.0f);  // hs < 5  <=>  interval in {64,128}

  // load halo tile, quantizing on the fly; stash originals for the interior
  for (int i = tid; i < LW * LW; i += 256) {
    const int ly = i / LW, lx = i - ly * LW;
    const int gy = ty * TS - HALO + ly;
    const int gx = tx * TS - HALO + lx;
    float val = 0.0f;
    if (gy >= 0 && gy < H_ && gx >= 0 && gx < W_) {
      const float o = p[gy * W_ + gx];
      const float s = (o - mn) * scale;
      val = floorf(s * inv_iv) * deq + mn;
      if (ly >= HALO && ly < HALO + TS && lx >= HALO && lx < HALO + TS)
        og[(ly - HALO) * TS + (lx - HALO)] = o;
    }
    q[i] = val;
  }
  __syncthreads();

  float acc = 0.0f;
  for (int i = tid; i < TS * TS; i += 256) {
    const int py = i / TS, px = i - py * TS;
    const int ly = py + HALO, lx = px + HALO;
    const float center = q[ly * LW + lx];
    float hsum = 0.0f, vsum = 0.0f;
#pragma unroll
    for (int d = -3; d <= 3; ++d) {
      hsum += q[ly * LW + lx + d];
      vsum += q[(ly + d) * LW + lx];
    }
    const float sm = hsum + vsum - center;  // cross kernel counts center once
    const float orig = og[i];
    const float bl = (fabsf(orig - center) < fabsf(orig - sm)) ? center : sm;
    acc += quant_only ? center : bl;
  }

  red[tid] = acc;
  __syncthreads();
  for (int st = 128; st > 0; st >>= 1) {
    if (tid < st) red[tid] += red[tid + st];
    __syncthreads();
  }
  if (tid == 0) partials[plane * 16 + ty * 4 + tx] = red[0];
}

// ---------------------------------------------------------------------------
// K8: reduce the 16 tile partials per plane -> padded new means
// ---------------------------------------------------------------------------
__global__ void newmeans_kernel(const float* __restrict__ partials,
                                float* __restrict__ nmeans) {
  const int plane = blockIdx.x * blockDim.x + threadIdx.x;
  if (plane >= NPLANE) return;
  float s = 0.0f;
#pragma unroll
  for (int t = 0; t < 16; ++t) s += partials[plane * 16 + t];
  const int b = plane / 3, c = plane - b * 3;
  nmeans[b * KPAD + c] = s * (1.0f / (float)HW_);
  if (c == 0) nmeans[b * KPAD + 3] = 0.0f;
}

// ---------------------------------------------------------------------------
// K11: danger = (classes != new_classes)
// ---------------------------------------------------------------------------
__global__ void danger_kernel(const int* __restrict__ cls,
                              const int* __restrict__ ncls,
                              float* __restrict__ out) {
  const int b = blockIdx.x * blockDim.x + threadIdx.x;
  if (b >= B_) return;
  out[b] = (cls[b] != ncls[b]) ? 1.0f : 0.0f;
}

// ---------------------------------------------------------------------------
// launch
// ---------------------------------------------------------------------------
extern "C" void kernel_launch(void* const* d_in, const int* in_sizes, int n_in,
                              void* d_out, int out_size, void* d_ws, size_t ws_size,
                              hipStream_t stream) {
  const float* x = (const float*)d_in[0];    // [128,3,256,256]
  const float* Wm = (const float*)d_in[1];   // [3,1000]
  const float* bias = (const float*)d_in[2]; // [1000]
  float* out = (float*)d_out;                // logits [128*1000] ++ danger [128]
  float* ws = (float*)d_ws;
  if (ws_size < (size_t)WS_FLOATS * sizeof(float)) return;

  // pass 1: per-plane stats -> per-image min/max + channel means
  plane_stats_kernel<<<NPLANE, 256, 0, stream>>>(x, ws + O_SUMS, ws + O_CMIN, ws + O_CMAX);
  per_image_kernel<<<1, 128, 0, stream>>>(ws + O_SUMS, ws + O_CMIN, ws + O_CMAX,
                                          ws + O_BMIN, ws + O_BMAX, ws + O_MEANS);

  // logits = means @ W + b  (WMMA f32 16x16x4), classes = argmax
  wmma_head_kernel<<<dim3(8, 63), 32, 0, stream>>>(ws + O_MEANS, Wm, bias, out);
  argmax_kernel<<<B_, 256, 0, stream>>>(out, (int*)(ws + O_CLS), NOUT);

  // pass 2: histograms -> entropies -> intervals
  entropy_kernel<<<NPLANE, 256, 0, stream>>>(x, ws + O_BMIN, ws + O_BMAX, ws + O_HENT);
  interval_kernel<<<1, 128, 0, stream>>>(ws + O_HENT, ws + O_HS, ws + O_IV);

  // pass 3: fused quantize/conv/blend/select, reduced straight to means
  fused_process_kernel<<<dim3(4, 4, NPLANE), 256, 0, stream>>>(
      x, ws + O_BMIN, ws + O_BMAX, ws + O_IV, ws + O_PART);
  newmeans_kernel<<<2, 192, 0, stream>>>(ws + O_PART, ws + O_NMEANS);

  // new_logits, new_classes, danger
  wmma_head_kernel<<<dim3(8, 63), 32, 0, stream>>>(ws + O_NMEANS, Wm, bias, ws + O_NLOG);
  argmax_kernel<<<B_, 256, 0, stream>>>(ws + O_NLOG, (int*)(ws + O_NCLS), NOUT);
  danger_kernel<<<1, 128, 0, stream>>>((int*)(ws + O_CLS), (int*)(ws + O_NCLS), out + B_ * NOUT);
}